// AdaptiveEdgeDropping_9775345566022
// MI455X (gfx1250) — compile-verified
//
#include <hip/hip_runtime.h>

typedef unsigned int u32;

// ---------------- Threefry2x32, key = jax.random.key(42) -> (0, 42) ----------------
__device__ __forceinline__ void threefry2x32_k42(u32 c0, u32 c1, u32& o0, u32& o1) {
  const u32 ks0 = 0u;
  const u32 ks1 = 42u;
  const u32 ks2 = 0x1BD11BDAu ^ ks0 ^ ks1;
  u32 x0 = c0 + ks0;
  u32 x1 = c1 + ks1;
#define TF_ROUND(r) { x0 += x1; x1 = (x1 << (r)) | (x1 >> (32 - (r))); x1 ^= x0; }
  TF_ROUND(13) TF_ROUND(15) TF_ROUND(26) TF_ROUND(6)
  x0 += ks1; x1 += ks2 + 1u;
  TF_ROUND(17) TF_ROUND(29) TF_ROUND(16) TF_ROUND(24)
  x0 += ks2; x1 += ks0 + 2u;
  TF_ROUND(13) TF_ROUND(15) TF_ROUND(26) TF_ROUND(6)
  x0 += ks0; x1 += ks1 + 3u;
  TF_ROUND(17) TF_ROUND(29) TF_ROUND(16) TF_ROUND(24)
  x0 += ks1; x1 += ks2 + 4u;
  TF_ROUND(13) TF_ROUND(15) TF_ROUND(26) TF_ROUND(6)
  x0 += ks2; x1 += ks0 + 5u;
#undef TF_ROUND
  o0 = x0; o1 = x1;
}

// key_i = log_sigmoid(dp*(1-v) - g*v) + Gumbel(uniform(bits)), mapped to an
// order-preserving uint32 so radix-select on uints == top-k on floats.
__device__ __forceinline__ u32 make_key(float v, u32 bits, float dp, float g) {
  float logits = fmaf(-(dp + g), v, dp);                 // dp*(1-v) - g*v
  float logp = (logits >= 0.0f) ? -log1pf(__expf(-logits))
                                : (logits - log1pf(__expf(logits)));
  float f = __uint_as_float((bits >> 9) | 0x3F800000u) - 1.0f;   // [0,1), JAX bit recipe
  float u = fmaxf(fmaf(f, 0.99999976f, 1e-7f), 1e-7f);           // minval/maxval clamp
  float gum = -__logf(-__logf(u));
  float key = logp + gum;
  u32 kb = __float_as_uint(key);
  u32 uk = (kb & 0x80000000u) ? ~kb : (kb | 0x80000000u);
  return (v != 0.0f) ? uk : 0u;                          // non-edges ineligible
}

#define NBIN0 2048            // radix widths: 11 / 11 / 10 bits
#define CTRL_OFF 2048         // ws: u32 hist[2048]; u32 ctrl[4]@2048 = {prefix,krem,T,pad}

__global__ __launch_bounds__(256) void init_kernel(u32* ws, const int* pct, int E) {
  int t = threadIdx.x;
  for (int i = t; i < NBIN0; i += 256) ws[i] = 0u;
  if (t == 0) {
    unsigned long long k =
        ((unsigned long long)(u32)E * (unsigned long long)(u32)(*pct)) / 100ull;
    u32* ctrl = ws + CTRL_OFF;
    ctrl[0] = 0u; ctrl[1] = (u32)k; ctrl[2] = 0u; ctrl[3] = 0u;
  }
}

// Fused: compute keys (exact JAX Threefry pairing: element i<E/2 uses counter
// pair (i, i+E/2) out0; element i+E/2 uses out1) + top-11-bit histogram.
__global__ __launch_bounds__(256) void keygen_hist(const float* __restrict__ mat,
                                                   const float* __restrict__ dparam,
                                                   const float* __restrict__ gparam,
                                                   u32* __restrict__ keys,
                                                   u32* __restrict__ ws, int E) {
  __shared__ u32 shist[NBIN0];
  for (int i = threadIdx.x; i < NBIN0; i += 256) shist[i] = 0u;
  __syncthreads();
  const float dp = dparam[0];
  const float g  = gparam[0];
  const int halfE = E >> 1;
  const int half4 = E >> 3;          // float4 count per half
  const float4* m4 = (const float4*)mat;
  uint4* k4 = (uint4*)keys;
  int gtid = blockIdx.x * blockDim.x + threadIdx.x;
  int gstride = gridDim.x * blockDim.x;
  for (int j = gtid; j < half4; j += gstride) {
    if (j + gstride < half4) {
      __builtin_prefetch(&m4[j + gstride], 0, 0);          // global_prefetch_b8
      __builtin_prefetch(&m4[j + gstride + half4], 0, 0);
    }
    float4 vlo = m4[j];
    float4 vhi = m4[j + half4];
    const float* plo = (const float*)&vlo;
    const float* phi = (const float*)&vhi;
    int base = j << 2;
    u32 lo[4], hi[4];
#pragma unroll
    for (int t = 0; t < 4; ++t) {
      u32 o0, o1;
      threefry2x32_k42((u32)(base + t), (u32)(base + t + halfE), o0, o1);
      lo[t] = make_key(plo[t], o0, dp, g);
      hi[t] = make_key(phi[t], o1, dp, g);
    }
    k4[j]         = make_uint4(lo[0], lo[1], lo[2], lo[3]);
    k4[j + half4] = make_uint4(hi[0], hi[1], hi[2], hi[3]);
#pragma unroll
    for (int t = 0; t < 4; ++t) {
      atomicAdd(&shist[lo[t] >> 21], 1u);
      atomicAdd(&shist[hi[t] >> 21], 1u);
    }
  }
  __syncthreads();
  for (int i = threadIdx.x; i < NBIN0; i += 256) {
    u32 v = shist[i];
    if (v) atomicAdd(&ws[i], v);
  }
}

// Radix histogram pass streamed through LDS with the gfx1250 async-to-LDS
// engine: per-wave double-buffered 512B chunks (ASYNCcnt is per-wave under
// wave32, so no cross-wave synchronization is needed).
__global__ __launch_bounds__(256) void hist_pass(const u32* __restrict__ keys,
                                                 u32* __restrict__ ws,
                                                 int shift, u32 himask, u32 binmask,
                                                 int E) {
  __shared__ uint4 stage[8][2][32];   // 8 waves x 2 buffers x 32 lanes x 16B
  __shared__ u32 shist[NBIN0];
  const u32* ctrl = ws + CTRL_OFF;
  int tid  = threadIdx.x;
  int lane = tid & 31;
  int w    = tid >> 5;
  for (int i = tid; i < NBIN0; i += 256) shist[i] = 0u;
  __syncthreads();

  const u32 pfx = ctrl[0] & himask;

  const int nChunks = E >> 7;                    // 128 keys (512B) per chunk
  int c = blockIdx.x * 8 + w;
  const int stride = gridDim.x * 8;

  unsigned lds0 = (unsigned)(size_t)&stage[w][0][lane];
  unsigned lds1 = (unsigned)(size_t)&stage[w][1][lane];

  int buf = 0;
  if (c < nChunks) {
    unsigned goff = (unsigned)c * 512u + (unsigned)lane * 16u;
    asm volatile("global_load_async_to_lds_b128 %0, %1, %2"
                 :: "v"(lds0), "v"(goff), "s"(keys) : "memory");
  }
  while (c < nChunks) {
    int cn = c + stride;                         // wave-uniform
    if (cn < nChunks) {
      unsigned goff = (unsigned)cn * 512u + (unsigned)lane * 16u;
      unsigned ldsn = buf ? lds0 : lds1;
      asm volatile("global_load_async_to_lds_b128 %0, %1, %2"
                   :: "v"(ldsn), "v"(goff), "s"(keys) : "memory");
      asm volatile("s_wait_asynccnt 0x1" ::: "memory");    // front buffer landed
    } else {
      asm volatile("s_wait_asynccnt 0x0" ::: "memory");
    }
    uint4 kv = stage[w][buf][lane];
    if ((kv.x & himask) == pfx) atomicAdd(&shist[(kv.x >> shift) & binmask], 1u);
    if ((kv.y & himask) == pfx) atomicAdd(&shist[(kv.y >> shift) & binmask], 1u);
    if ((kv.z & himask) == pfx) atomicAdd(&shist[(kv.z >> shift) & binmask], 1u);
    if ((kv.w & himask) == pfx) atomicAdd(&shist[(kv.w >> shift) & binmask], 1u);
    buf ^= 1;
    c = cn;
  }
  __syncthreads();
  for (int i = tid; i < NBIN0; i += 256) {
    u32 v = shist[i];
    if (v) atomicAdd(&ws[i], v);
  }
}

// Find the bucket containing the k-th largest key. Two-level: 256 threads
// reduce groups of (nbins/256) bins, thread 0 scans 256 group sums + <=8 bins.
__global__ __launch_bounds__(256) void scan_kernel(u32* ws, int shift, int nbins) {
  __shared__ u32 gsum[256];
  u32* ctrl = ws + CTRL_OFF;
  int t = threadIdx.x;
  const int gsize = nbins >> 8;                  // 8 (2048 bins) or 4 (1024 bins)
  u32 s = 0;
  for (int i = 0; i < 8; ++i)
    if (i < gsize) s += ws[t * gsize + i];
  gsum[t] = s;
  __syncthreads();
  if (t == 0) {
    u32 krem = ctrl[1];
    u32 cum = 0;
    int g = 255;
    for (; g > 0; --g) {
      if (cum + gsum[g] >= krem) break;
      cum += gsum[g];
    }
    int b = g * gsize + gsize - 1;
    for (; b > g * gsize; --b) {
      u32 h = ws[b];
      if (cum + h >= krem) break;
      cum += h;
    }
    ctrl[0] |= ((u32)b << shift);
    ctrl[1] = krem - cum;
    if (shift == 0) ctrl[2] = ctrl[0];           // final threshold T (k-th largest)
  }
  __syncthreads();
  for (int i = t; i < nbins; i += 256) ws[i] = 0u;   // clear for next pass
}

__global__ __launch_bounds__(256) void finalize_kernel(const float* __restrict__ mat,
                                                       u32* keys, float* out,
                                                       const u32* __restrict__ ws, int E) {
  const u32 T = ws[CTRL_OFF + 2];
  int n4 = E >> 2;
  const float4* m4 = (const float4*)mat;
  uint4* k4 = (uint4*)keys;
  float4* o4 = (float4*)out;
  for (int i = blockIdx.x * blockDim.x + threadIdx.x; i < n4;
       i += gridDim.x * blockDim.x) {
    uint4 kv = k4[i];                    // read before aliased write
    float4 mv = m4[i];
    float4 r;
    r.x = (kv.x >= T) ? 0.0f : mv.x;
    r.y = (kv.y >= T) ? 0.0f : mv.y;
    r.z = (kv.z >= T) ? 0.0f : mv.z;
    r.w = (kv.w >= T) ? 0.0f : mv.w;
    o4[i] = r;
  }
}

extern "C" void kernel_launch(void* const* d_in, const int* in_sizes, int n_in,
                              void* d_out, int out_size, void* d_ws, size_t ws_size,
                              hipStream_t stream) {
  const float* mat = (const float*)d_in[0];
  const float* dp  = (const float*)d_in[1];
  const float* gm  = (const float*)d_in[2];
  const int*   pct = (const int*)d_in[3];
  const int E = in_sizes[0];             // 4096*4096, multiple of 128

  u32*   ws   = (u32*)d_ws;              // 2048 hist + 4 ctrl words (~8.2 KB)
  u32*   keys = (u32*)d_out;             // d_out doubles as the key buffer
  float* out  = (float*)d_out;

  init_kernel<<<1, 256, 0, stream>>>(ws, pct, E);

  // Pass 0 (bits 31:21) fused with key generation.
  keygen_hist<<<4096, 256, 0, stream>>>(mat, dp, gm, keys, ws, E);
  scan_kernel<<<1, 256, 0, stream>>>(ws, 21, 2048);

  // Pass 1 (bits 20:10), pass 2 (bits 9:0) — async-to-LDS streamed.
  hist_pass<<<1024, 256, 0, stream>>>(keys, ws, 10, 0xFFE00000u, 2047u, E);
  scan_kernel<<<1, 256, 0, stream>>>(ws, 10, 2048);
  hist_pass<<<1024, 256, 0, stream>>>(keys, ws, 0, 0xFFFFFC00u, 1023u, E);
  scan_kernel<<<1, 256, 0, stream>>>(ws, 0, 1024);

  finalize_kernel<<<4096, 256, 0, stream>>>(mat, keys, out, ws, E);
}